// DifferentiableGaussianRenderer_65085934403899
// MI455X (gfx1250) — compile-verified
//
#include <hip/hip_runtime.h>
#include <math.h>

#define GH 192
#define GW 256
#define NG 512
#define SLOT 12   // floats per sorted gaussian: ca,cb,cc,mx,my,pad,pad,pad,op,r,g,b

typedef __attribute__((ext_vector_type(2))) float v2f;
typedef __attribute__((ext_vector_type(8))) float v8f;
typedef __attribute__((__vector_size__(16))) int v4i;

#ifndef __has_builtin
#define __has_builtin(x) 0
#endif

#if __has_builtin(__builtin_amdgcn_global_load_async_to_lds_b128)
#define HAVE_ASYNC_LDS 1
#else
#define HAVE_ASYNC_LDS 0
#endif

// ---------------------------------------------------------------------------
// Kernel 1: per-gaussian preprocessing + bitonic depth sort (1 block x 512)
// ---------------------------------------------------------------------------
__global__ __launch_bounds__(NG) void gs_preprocess_sort(
    const float* __restrict__ pos, const float* __restrict__ scl,
    const float* __restrict__ rot, const float* __restrict__ opa,
    const float* __restrict__ col, const float* __restrict__ vm,
    const int* __restrict__ fovp, float* __restrict__ params)
{
    __shared__ float s_ca[NG], s_cb[NG], s_cc[NG], s_mx[NG], s_my[NG];
    __shared__ float s_op[NG], s_r[NG], s_g[NG], s_b[NG];
    __shared__ float s_key[NG];
    __shared__ int   s_idx[NG];
    const int i = threadIdx.x;

    {
        // view matrix rows (row-major 4x4): Wr = vm[:3,:3], t = vm[:3,3]
        float W00 = vm[0], W01 = vm[1], W02 = vm[2],  t0 = vm[3];
        float W10 = vm[4], W11 = vm[5], W12 = vm[6],  t1 = vm[7];
        float W20 = vm[8], W21 = vm[9], W22 = vm[10], t2 = vm[11];
        double fov = (double)fovp[0];
        float fx = (float)((double)GW / (2.0 * tan(fov * 3.14159265358979323846 / 360.0)));

        // quaternion -> rotation
        float qw = rot[4*i+0], qx = rot[4*i+1], qy = rot[4*i+2], qz = rot[4*i+3];
        float inv = 1.0f / sqrtf(qw*qw + qx*qx + qy*qy + qz*qz);
        qw *= inv; qx *= inv; qy *= inv; qz *= inv;
        float R00 = 1.0f-2.0f*(qy*qy+qz*qz), R01 = 2.0f*(qx*qy-qw*qz), R02 = 2.0f*(qx*qz+qw*qy);
        float R10 = 2.0f*(qx*qy+qw*qz), R11 = 1.0f-2.0f*(qx*qx+qz*qz), R12 = 2.0f*(qy*qz-qw*qx);
        float R20 = 2.0f*(qx*qz-qw*qy), R21 = 2.0f*(qy*qz+qw*qx), R22 = 1.0f-2.0f*(qx*qx+qy*qy);

        float s0 = scl[3*i+0], s1 = scl[3*i+1], s2 = scl[3*i+2];
        float e0 = s0*s0, e1 = s1*s1, e2 = s2*s2;
        // cov3d = R diag(e) R^T (symmetric)
        float C00 = R00*R00*e0 + R01*R01*e1 + R02*R02*e2;
        float C01 = R00*R10*e0 + R01*R11*e1 + R02*R12*e2;
        float C02 = R00*R20*e0 + R01*R21*e1 + R02*R22*e2;
        float C11 = R10*R10*e0 + R11*R11*e1 + R12*R12*e2;
        float C12 = R10*R20*e0 + R11*R21*e1 + R12*R22*e2;
        float C22 = R20*R20*e0 + R21*R21*e1 + R22*R22*e2;

        float px = pos[3*i+0], py = pos[3*i+1], pz = pos[3*i+2];
        float X = W00*px + W01*py + W02*pz + t0;
        float Y = W10*px + W11*py + W12*pz + t1;
        float Z = W20*px + W21*py + W22*pz + t2;
        float depth = Z;
        float zs = fmaxf(depth, 0.1f);
        float mx = fx * X / zs + 0.5f * (float)GW;
        float my = 0.5f * (float)GH - fx * Y / zs;

        // J (2x3), T = J @ Wr
        float j00 = fx / zs;
        float j02 = -fx * X / (zs*zs);
        float j12 = -fx * Y / (zs*zs);
        float T00 = j00*W00 + j02*W20, T01 = j00*W01 + j02*W21, T02 = j00*W02 + j02*W22;
        float T10 = j00*W10 + j12*W20, T11 = j00*W11 + j12*W21, T12 = j00*W12 + j12*W22;

        // cov2d = T C T^T
        float u0 = C00*T00 + C01*T01 + C02*T02;
        float u1 = C01*T00 + C11*T01 + C12*T02;
        float u2 = C02*T00 + C12*T01 + C22*T02;
        float v0 = C00*T10 + C01*T11 + C02*T12;
        float v1 = C01*T10 + C11*T11 + C12*T12;
        float v2 = C02*T10 + C12*T11 + C22*T12;
        float a = T00*u0 + T01*u1 + T02*u2;
        float b = T10*u0 + T11*u1 + T12*u2;
        float c = T10*v0 + T11*v1 + T12*v2;

        float mean_e = 0.5f * (a + c);
        float disc = sqrtf(fmaxf(0.25f*(a-c)*(a-c) + b*b, 0.0f));
        float min_eig = mean_e - disc;
        float eps = (min_eig <= 0.0f) ? (fabsf(min_eig) + 1e-6f) : 0.0f;
        a += eps; c += eps;
        float max_eig = mean_e + eps + disc;
        float radii = ceilf(3.0f * sqrtf(fmaxf(max_eig, 1e-6f)));
        bool visible = (depth > 0.1f) && (depth < 100.0f) && (radii > 0.0f);

        float det = fmaxf(a*c - b*b, 1e-12f);
        float idet = 1.0f / det;
        s_ca[i] = c * idet;
        s_cb[i] = -b * idet;
        s_cc[i] = a * idet;
        s_mx[i] = mx;
        s_my[i] = my;
        float sg = 1.0f / (1.0f + expf(-opa[i]));
        s_op[i] = visible ? sg : 0.0f;     // fold visibility into opacity (exact)
        s_r[i] = col[3*i+0]; s_g[i] = col[3*i+1]; s_b[i] = col[3*i+2];
        s_key[i] = visible ? -depth : INFINITY;
        s_idx[i] = i;
    }
    __syncthreads();

    // Bitonic sort ascending on (key, idx) -- matches stable argsort(-depth|inf)
    for (int k = 2; k <= NG; k <<= 1) {
        for (int j = k >> 1; j > 0; j >>= 1) {
            int ixj = i ^ j;
            if (ixj > i) {
                float ki = s_key[i], kj = s_key[ixj];
                int   ai = s_idx[i], aj = s_idx[ixj];
                bool up = ((i & k) == 0);
                bool gt = (ki > kj) || ((ki == kj) && (ai > aj));
                if (gt == up) {
                    s_key[i] = kj; s_key[ixj] = ki;
                    s_idx[i] = aj; s_idx[ixj] = ai;
                }
            }
            __syncthreads();
        }
    }

    int g = s_idx[i];
    float* o = params + i * SLOT;
    o[0] = s_ca[g]; o[1] = s_cb[g]; o[2] = s_cc[g]; o[3] = s_mx[g]; o[4] = s_my[g];
    o[5] = 0.0f; o[6] = 0.0f; o[7] = 0.0f;
    o[8] = s_op[g]; o[9] = s_r[g]; o[10] = s_g[g]; o[11] = s_b[g];
}

// ---------------------------------------------------------------------------
// Kernel 2: per-pixel alpha compositing. One block per row (256 px = 8 waves
// x 32 px). Quadratic q(xl) per 16-gaussian chunk evaluated on the matrix
// pipe via V_WMMA_F32_16X16X4_F32 (K=3 basis: xl^2, xl, 1). Parameter table
// staged into LDS with GLOBAL_LOAD_ASYNC_TO_LDS_B128 when available.
// ---------------------------------------------------------------------------
__global__ __launch_bounds__(256) void gs_render(
    const float* __restrict__ params, const float* __restrict__ bg,
    float* __restrict__ out)
{
    __shared__ float4 spv[NG * SLOT / 4];
    float* sp = (float*)spv;
    const int tid = threadIdx.x;

#if HAVE_ASYNC_LDS
    {
        // Builtin signature (from compiler diagnostic): arg0 is a non-const
        // global pointer to a 16-byte int vector; arg1 the LDS counterpart.
        __attribute__((address_space(1))) v4i* g =
            (__attribute__((address_space(1))) v4i*)params;
        __attribute__((address_space(3))) v4i* l =
            (__attribute__((address_space(3))) v4i*)sp;
        #pragma unroll
        for (int k = 0; k < (NG * SLOT * 4) / (256 * 16); ++k) {
            int idx = tid + k * 256;     // in 16-byte units
            __builtin_amdgcn_global_load_async_to_lds_b128(g + idx, l + idx, 0, 0);
        }
#if __has_builtin(__builtin_amdgcn_s_wait_asynccnt)
        __builtin_amdgcn_s_wait_asynccnt(0);
#else
        asm volatile("s_wait_asynccnt 0" ::: "memory");
#endif
    }
#else
    {
        const float4* src = (const float4*)params;
        #pragma unroll
        for (int k = 0; k < (NG * SLOT / 4) / 256; ++k)
            spv[tid + k * 256] = src[tid + k * 256];
    }
#endif
    __syncthreads();

    const int lane  = tid & 31;
    const int wave  = tid >> 5;
    const int y     = blockIdx.x;
    const int xbase = wave * 32;
    const int nloc  = lane & 15;
    const bool lowHalf = (lane < 16);
    const float yf = (float)y;
    const float x0 = (float)xbase;

    // B matrices (constant per lane): basis rows (xl^2, xl, 1, 0) for 16 px.
    // B 4x16 layout: lanes0-15 hold K=0(v0),K=1(v1); lanes16-31 K=2(v0),K=3(v1).
    float xl1 = (float)nloc;
    float xl2 = (float)(nloc + 16);
    v2f B1, B2;
    B1.x = lowHalf ? xl1 * xl1 : 1.0f;
    B1.y = lowHalf ? xl1       : 0.0f;
    B2.x = lowHalf ? xl2 * xl2 : 1.0f;
    B2.y = lowHalf ? xl2       : 0.0f;
    const v8f zeroC = {0.f, 0.f, 0.f, 0.f, 0.f, 0.f, 0.f, 0.f};

    float T = 1.0f, ar = 0.0f, ag = 0.0f, ab = 0.0f;

    for (int cbase = 0; cbase < NG; cbase += 16) {
        // A 16x4 layout: lane m<16 holds A[m][0],A[m][1]; lane m+16 holds A[m][2],A[m][3]
        const float* gp = sp + (cbase + nloc) * SLOT;
        float ca = gp[0], cb = gp[1], cc = gp[2], mx = gp[3], my = gp[4];
        float dxc = x0 - mx;
        float dy  = yf - my;
        float lin = 2.0f * (ca * dxc + cb * dy);
        float q0  = (ca * dxc + 2.0f * cb * dy) * dxc + cc * dy * dy;
        v2f A;
        A.x = lowHalf ? ca  : q0;
        A.y = lowHalf ? lin : 0.0f;

        v8f D1 = __builtin_amdgcn_wmma_f32_16x16x4_f32(
            false, A, false, B1, (short)0, zeroC, false, false);
        v8f D2 = __builtin_amdgcn_wmma_f32_16x16x4_f32(
            false, A, false, B2, (short)0, zeroC, false, false);

        // D layout: lanes0-15 hold M=0..7 (pixel n=lane), lanes16-31 hold M=8..15.
        // Each lane only needs 8 values from its partner: lanes<16 need the
        // partner's D1 regs, lanes>=16 need the partner's D2 regs. Select first,
        // then do a single xor-16 exchange per register pair (8 bpermutes/chunk).
        float lo[8], hi[8];
        #pragma unroll
        for (int jj = 0; jj < 8; ++jj) {
            float d1 = D1[jj], d2 = D2[jj];
            float s  = lowHalf ? d2 : d1;        // what the partner lane needs
            float o  = __shfl_xor(s, 16, 32);    // receive partner's selection
            lo[jj] = lowHalf ? d1 : o;           // gaussians cbase+0..7
            hi[jj] = lowHalf ? o  : d2;          // gaussians cbase+8..15
        }

        const float* cp = sp + cbase * SLOT;
        #pragma unroll
        for (int mm = 0; mm < 16; ++mm) {
            float qv = (mm < 8) ? lo[mm] : hi[mm - 8];
            const float4 oc = *(const float4*)(cp + mm * SLOT + 8); // op,r,g,b
            float G = __expf(-0.5f * fmaxf(qv, 0.0f));
            float alpha = fminf(oc.x * G, 0.99f);
            float w = T * alpha;
            ar = fmaf(w, oc.y, ar);
            ag = fmaf(w, oc.z, ag);
            ab = fmaf(w, oc.w, ab);
            T *= (1.0f - alpha);
        }
    }

    const int px = xbase + lane;
    float* o = out + (y * GW + px) * 3;
    o[0] = ar + T * bg[0];
    o[1] = ag + T * bg[1];
    o[2] = ab + T * bg[2];
}

// ---------------------------------------------------------------------------
extern "C" void kernel_launch(void* const* d_in, const int* in_sizes, int n_in,
                              void* d_out, int out_size, void* d_ws, size_t ws_size,
                              hipStream_t stream) {
    (void)in_sizes; (void)n_in; (void)out_size; (void)ws_size;
    const float* pos = (const float*)d_in[0];
    const float* scl = (const float*)d_in[1];
    const float* rot = (const float*)d_in[2];
    const float* opa = (const float*)d_in[3];
    const float* col = (const float*)d_in[4];
    const float* vm  = (const float*)d_in[5];
    const float* bg  = (const float*)d_in[6];
    const int*   fov = (const int*)d_in[7];
    float* params = (float*)d_ws;   // needs NG*SLOT*4 = 24576 bytes

    gs_preprocess_sort<<<1, NG, 0, stream>>>(pos, scl, rot, opa, col, vm, fov, params);
    gs_render<<<GH, 256, 0, stream>>>(params, bg, (float*)d_out);
}